// TriangleScene_77764677861376
// MI455X (gfx1250) — compile-verified
//
#include <hip/hip_runtime.h>
#include <hip/hip_bf16.h>
#include <stdint.h>

// ---------------------------------------------------------------------------
// CDNA5 / gfx1250 types
// ---------------------------------------------------------------------------
typedef float v2f  __attribute__((ext_vector_type(2)));
typedef float v8f  __attribute__((ext_vector_type(8)));
typedef unsigned int u32x4 __attribute__((ext_vector_type(4)));
typedef int  i32x4 __attribute__((ext_vector_type(4)));
typedef int  i32x8 __attribute__((ext_vector_type(8)));

#define EPS_F      1e-6f
#define HIT_THR    0.999f      // 1.0 - HIT_TOL
#define MAXTP      256         // max padded triangle count (T=96 here)

// ---------------------------------------------------------------------------
// small float3 helpers
// ---------------------------------------------------------------------------
struct f3 { float x, y, z; };
static __device__ __forceinline__ f3 ld3(const float* p) { return f3{p[0], p[1], p[2]}; }
static __device__ __forceinline__ f3 add3(f3 a, f3 b) { return f3{a.x+b.x, a.y+b.y, a.z+b.z}; }
static __device__ __forceinline__ f3 sub3(f3 a, f3 b) { return f3{a.x-b.x, a.y-b.y, a.z-b.z}; }
static __device__ __forceinline__ f3 scl3(f3 a, float s) { return f3{a.x*s, a.y*s, a.z*s}; }
static __device__ __forceinline__ float dot3(f3 a, f3 b) { return a.x*b.x + a.y*b.y + a.z*b.z; }
static __device__ __forceinline__ f3 cross3(f3 a, f3 b) {
    return f3{a.y*b.z - a.z*b.y, a.z*b.x - a.x*b.z, a.x*b.y - a.y*b.x};
}

// ---------------------------------------------------------------------------
// WMMA f32 16x16x4 wrapper (D = A(16x4) * B(4x16) + 0)
// A layout: lane<16 -> {K0,K1} of row M=lane ; lane>=16 -> {K2,K3} of row M=lane-16
// B layout: lane<16 -> {K0,K1} of col N=lane ; lane>=16 -> {K2,K3} of col N=lane-16
// D layout: VGPR k, lane<16 -> (M=k,   N=lane) ; lane>=16 -> (M=k+8, N=lane-16)
// ---------------------------------------------------------------------------
static __device__ __forceinline__ v8f wmma4(v2f a, v2f b) {
    v8f c = {0.f, 0.f, 0.f, 0.f, 0.f, 0.f, 0.f, 0.f};
    return __builtin_amdgcn_wmma_f32_16x16x4_f32(false, a, false, b,
                                                 (short)0, c, false, false);
}

// ---------------------------------------------------------------------------
// Tensor Data Mover: stage `ndwords` DWORDs from global -> LDS offset 0.
// Descriptor built per CDNA5 ISA §8 (1D tile: tile_dim1 = 0 => unused).
// ---------------------------------------------------------------------------
static __device__ __forceinline__ void tdm_load_to_lds(const void* gptr,
                                                       unsigned ldsOff,
                                                       unsigned ndwords) {
    uint64_t ga = (uint64_t)gptr;
    u32x4 g0;
    g0[0] = 1u;                                     // count=1 (valid), user mode
    g0[1] = ldsOff;                                 // lds_addr (bytes)
    g0[2] = (unsigned)(ga & 0xFFFFFFFFu);           // global_addr[31:0]
    g0[3] = (unsigned)((ga >> 32) & 0x01FFFFFFu)    // global_addr[56:32]
          | 0x80000000u;                            // type = 2 ("image")
    i32x8 g1;
    g1[0] = (int)(2u << 16);                        // data_size = 4 bytes
    g1[1] = (int)((ndwords & 0xFFFFu) << 16);       // tensor_dim0[15:0]
    g1[2] = (int)(((ndwords >> 16) & 0xFFFFu)       // tensor_dim0[31:16]
          | (1u << 16));                            // tensor_dim1 = 1
    g1[3] = (int)((ndwords & 0xFFFFu) << 16);       // tile_dim0 = ndwords
    g1[4] = 0;                                      // tile_dim1/2 = 0 (unused)
    g1[5] = (int)ndwords;                           // tensor_dim0_stride
    g1[6] = 0;
    g1[7] = 0;
    i32x4 z4 = {0, 0, 0, 0};
#if defined(__clang_major__) && (__clang_major__ >= 23)
    i32x8 z8 = {0, 0, 0, 0, 0, 0, 0, 0};
    __builtin_amdgcn_tensor_load_to_lds(g0, g1, z4, z4, z8, 0);
#else
    __builtin_amdgcn_tensor_load_to_lds(g0, g1, z4, z4, 0);
#endif
}

// ---------------------------------------------------------------------------
// Kernel 1: image-method paths, full vertices, objects, mask1&mask2
// one thread per (tx, rx, path) instance
// ---------------------------------------------------------------------------
__global__ void paths_kernel(const float* __restrict__ txv,
                             const float* __restrict__ rxv,
                             const float* __restrict__ verts,
                             const int*   __restrict__ tris,
                             const float* __restrict__ norms,
                             float* __restrict__ outFull,
                             float* __restrict__ outObj,
                             int*   __restrict__ mask12,
                             int T, int RX, int P, int N) {
    int q = blockIdx.x * blockDim.x + threadIdx.x;
    if (q >= N) return;
    __builtin_prefetch(verts, 0, 0);   // global_prefetch_b8 warm-up of geometry

    int tx  = q / (RX * P);
    int rem = q - tx * (RX * P);
    int rx  = rem / P;
    int p   = rem - rx * P;
    int i   = p / (T - 1);
    int j0  = p - i * (T - 1);
    int j   = j0 + (j0 >= i ? 1 : 0);

    f3 F = ld3(txv + 3 * tx);
    f3 R = ld3(rxv + 3 * rx);

    int ia = tris[3*i], ib = tris[3*i+1], ic = tris[3*i+2];
    int ja = tris[3*j], jb = tris[3*j+1], jc = tris[3*j+2];
    f3 A0 = ld3(verts + 3*ia), B0 = ld3(verts + 3*ib), C0 = ld3(verts + 3*ic);
    f3 A1 = ld3(verts + 3*ja), B1 = ld3(verts + 3*jb), C1 = ld3(verts + 3*jc);
    f3 n0 = ld3(norms + 3*i),  n1 = ld3(norms + 3*j);
    f3 v00 = A0, v01 = A1;                        // mirror vertices (first vertex)

    // forward scan: reflect across mirror 0 then mirror 1
    f3 img1 = sub3(F,    scl3(n0, 2.f * dot3(sub3(F,    v00), n0)));
    f3 img2 = sub3(img1, scl3(n1, 2.f * dot3(sub3(img1, v01), n1)));

    // backward scan (reverse): intersect mirrors (IEEE div: feeds `full` output)
    float den1 = dot3(sub3(R, img2), n1);
    float t1   = dot3(sub3(v01, img2), n1) / den1;
    f3 V1 = add3(img2, scl3(sub3(R, img2), t1));
    float den0 = dot3(sub3(V1, img1), n0);
    float t0   = dot3(sub3(v00, img1), n0) / den0;
    f3 V0 = add3(img1, scl3(sub3(V1, img1), t0));

    // mask1: contains tests
    f3 u0  = cross3(sub3(B0, A0), sub3(V0, A0));
    f3 w0v = cross3(sub3(C0, B0), sub3(V0, B0));
    f3 w0w = cross3(sub3(A0, C0), sub3(V0, C0));
    bool c0 = (dot3(u0, w0v) >= 0.f) && (dot3(w0v, w0w) >= 0.f) && (dot3(w0w, u0) >= 0.f);
    f3 u1  = cross3(sub3(B1, A1), sub3(V1, A1));
    f3 w1v = cross3(sub3(C1, B1), sub3(V1, B1));
    f3 w1w = cross3(sub3(A1, C1), sub3(V1, C1));
    bool c1 = (dot3(u1, w1v) >= 0.f) && (dot3(w1v, w1w) >= 0.f) && (dot3(w1w, u1) >= 0.f);

    // mask2: same-side tests (full = [F, V0, V1, R])
    float dprev0 = dot3(sub3(F,  v00), n0), dnext0 = dot3(sub3(V1, v00), n0);
    float dprev1 = dot3(sub3(V0, v01), n1), dnext1 = dot3(sub3(R,  v01), n1);
    bool m2 = (dprev0 * dnext0 >= 0.f) && (dprev1 * dnext1 >= 0.f);

    mask12[q] = (c0 && c1 && m2) ? 1 : 0;

    float* o = outFull + (size_t)q * 12;
    o[0]=F.x;  o[1]=F.y;  o[2]=F.z;
    o[3]=V0.x; o[4]=V0.y; o[5]=V0.z;
    o[6]=V1.x; o[7]=V1.y; o[8]=V1.z;
    o[9]=R.x;  o[10]=R.y; o[11]=R.z;

    float* ob = outObj + (size_t)q * 4;
    ob[0] = (float)tx; ob[1] = (float)i; ob[2] = (float)j; ob[3] = (float)rx;
}

// ---------------------------------------------------------------------------
// Kernel 2: per-triangle occlusion vectors (n, U, V) with folded constants.
// Column format float4 (x, y, z, -dot(vec, v0)) -> padded to TP (mult. of 16).
// ---------------------------------------------------------------------------
__global__ void trivec_kernel(const float* __restrict__ verts,
                              const int*   __restrict__ tris,
                              float4* __restrict__ triVec,
                              int T, int TP) {
    int j = blockIdx.x * blockDim.x + threadIdx.x;
    if (j >= TP) return;
    if (j >= T) {
        float4 z = {0.f, 0.f, 0.f, 0.f};
        triVec[3*j] = z; triVec[3*j+1] = z; triVec[3*j+2] = z;
        return;
    }
    f3 v0 = ld3(verts + 3 * tris[3*j]);
    f3 v1 = ld3(verts + 3 * tris[3*j+1]);
    f3 v2 = ld3(verts + 3 * tris[3*j+2]);
    f3 e1 = sub3(v1, v0), e2 = sub3(v2, v0);
    f3 n  = cross3(e1, e2);
    float nn  = dot3(n, n);
    float inv = (nn > 0.f) ? (1.f / nn) : 0.f;
    f3 U = scl3(cross3(e2, n), inv);              // U.e1=1, U.e2=0
    f3 V = scl3(cross3(n, e1), inv);              // V.e1=0, V.e2=1
    triVec[3*j+0] = float4{n.x, n.y, n.z, -dot3(n, v0)};
    triVec[3*j+1] = float4{U.x, U.y, U.z, -dot3(U, v0)};
    triVec[3*j+2] = float4{V.x, V.y, V.z, -dot3(V, v0)};
}

// ---------------------------------------------------------------------------
// Kernel 3: WMMA occlusion. One wave32 owns 16 ray segments; triangles
// batched 16 at a time from LDS (staged via TDM). 6 v_wmma_f32_16x16x4_f32
// per batch produce all o/d dot products; per-VGPR ballots reduce any-hit.
// ---------------------------------------------------------------------------
__global__ void occl_kernel(const float* __restrict__ outFull,
                            const float4* __restrict__ triVec,
                            int* __restrict__ segHit,
                            int S, int TP) {
    __shared__ float4 sTri[MAXTP * 3];

    int lane = threadIdx.x & 31;
    int wave = threadIdx.x >> 5;
    int nv   = TP * 3;

    // zero-init (pad region + makes LDS contents well-defined for the compiler)
    for (int k = threadIdx.x; k < nv; k += blockDim.x)
        sTri[k] = float4{0.f, 0.f, 0.f, 0.f};
    __syncthreads();

    if (wave == 0) {                       // one TDM issue per workgroup
        tdm_load_to_lds(triVec, 0u, (unsigned)(TP * 12));
        __builtin_amdgcn_s_wait_tensorcnt(0);
    }
    __syncthreads();
    // Compiler barrier only (no pointer laundering: keep LDS address space so
    // B-operand reads lower to ds_load_b128, not flat_load).
    asm volatile("" ::: "memory");

    int g       = blockIdx.x * (blockDim.x >> 5) + wave;
    int segBase = g * 16;
    int col     = lane & 15;
    int s       = segBase + col;

    float ox=0.f, oy=0.f, oz=0.f, dx=0.f, dy=0.f, dz=0.f;
    if (s < S) {
        int q = s / 3, k = s - 3 * q;
        const float* f = outFull + (size_t)q * 12 + (size_t)k * 3;
        ox = f[0]; oy = f[1]; oz = f[2];
        dx = f[3] - ox; dy = f[4] - oy; dz = f[5] - oz;
    }
    v2f Ao, Ad;
    if (lane < 16) { Ao.x = ox; Ao.y = oy;  Ad.x = dx; Ad.y = dy;  }
    else           { Ao.x = oz; Ao.y = 1.f; Ad.x = dz; Ad.y = 0.f; }

    unsigned acc[8] = {0u,0u,0u,0u,0u,0u,0u,0u};
    int nb = TP >> 4;
    for (int tb = 0; tb < nb; ++tb) {
        int tri = tb * 16 + col;
        float4 wn = sTri[tri*3+0];
        float4 wu = sTri[tri*3+1];
        float4 wv = sTri[tri*3+2];
        v2f Bn, Bu, Bv;
        if (lane < 16) { Bn.x=wn.x; Bn.y=wn.y; Bu.x=wu.x; Bu.y=wu.y; Bv.x=wv.x; Bv.y=wv.y; }
        else           { Bn.x=wn.z; Bn.y=wn.w; Bu.x=wu.z; Bu.y=wu.w; Bv.x=wv.z; Bv.y=wv.w; }

        v8f Don = wmma4(Ao, Bn);   // o.n - n.v0
        v8f Ddn = wmma4(Ad, Bn);   // d.n
        v8f Dou = wmma4(Ao, Bu);   // o.U - U.v0
        v8f Ddu = wmma4(Ad, Bu);   // d.U
        v8f Dov = wmma4(Ao, Bv);   // o.V - V.v0
        v8f Ddv = wmma4(Ad, Bv);   // d.V

        #pragma unroll
        for (int k = 0; k < 8; ++k) {
            float dn = Ddn[k];
            // boolean test with generous epsilons: v_rcp_f32 (1 ulp) suffices,
            // avoids the 15-op IEEE division expansion in the hot loop
            float t  = -Don[k] * __builtin_amdgcn_rcpf(dn);
            float u  = Dou[k] + t * Ddu[k];      // barycentric u (e1 weight)
            float v  = Dov[k] + t * Ddv[k];      // barycentric v (e2 weight)
            bool hit = (__builtin_fabsf(dn) > EPS_F) &&
                       (u >= 0.f) && (v >= 0.f) && ((u + v) <= 1.f) &&
                       (t > EPS_F) && (t < HIT_THR);
            acc[k] |= __builtin_amdgcn_ballot_w32(hit);
        }
    }

    if (lane < 16) {
        int sOut = segBase + lane;
        if (sOut < S) {
            unsigned a = acc[lane & 7];
            unsigned h = (lane < 8) ? (a & 0xFFFFu) : (a >> 16);
            segHit[sOut] = h ? 1 : 0;
        }
    }
}

// ---------------------------------------------------------------------------
// Kernel 4: final mask = mask1&mask2 & no-segment-occluded
// ---------------------------------------------------------------------------
__global__ void finalize_kernel(const int* __restrict__ mask12,
                                const int* __restrict__ segHit,
                                float* __restrict__ outMask, int N) {
    int q = blockIdx.x * blockDim.x + threadIdx.x;
    if (q >= N) return;
    bool m = mask12[q] && !segHit[3*q] && !segHit[3*q+1] && !segHit[3*q+2];
    outMask[q] = m ? 1.0f : 0.0f;
}

// ---------------------------------------------------------------------------
// Host-side launcher
// ---------------------------------------------------------------------------
extern "C" void kernel_launch(void* const* d_in, const int* in_sizes, int n_in,
                              void* d_out, int out_size, void* d_ws, size_t ws_size,
                              hipStream_t stream) {
    const float* txv   = (const float*)d_in[0];
    const float* rxv   = (const float*)d_in[1];
    const float* verts = (const float*)d_in[2];
    const int*   tris  = (const int*)  d_in[3];
    const float* norms = (const float*)d_in[4];
    // d_in[5] = order (always 2 for this problem; ORDER=2 path is hardcoded)

    const int TX = in_sizes[0] / 3;
    const int RX = in_sizes[1] / 3;
    const int T  = in_sizes[3] / 3;
    const int P  = T * (T - 1);
    const int N  = TX * RX * P;      // path instances
    const int S  = 3 * N;            // ray segments
    const int TP = (T + 15) & ~15;   // triangles padded to WMMA tile width

    // output layout: full (N*12) | objects (N*4) | mask (N)
    float* outFull = (float*)d_out;
    float* outObj  = outFull + (size_t)N * 12;
    float* outMask = outFull + (size_t)N * 16;

    // workspace layout: triVec (TP*3 float4, 16B aligned) | mask12 | segHit
    char* ws = (char*)d_ws;
    float4* triVec = (float4*)ws;
    int* mask12 = (int*)(ws + (size_t)TP * 3 * sizeof(float4));
    int* segHit = mask12 + N;

    const int BT = 256;
    paths_kernel<<<(N + BT - 1) / BT, BT, 0, stream>>>(
        txv, rxv, verts, tris, norms, outFull, outObj, mask12, T, RX, P, N);

    trivec_kernel<<<(TP + BT - 1) / BT, BT, 0, stream>>>(verts, tris, triVec, T, TP);

    const int groups = (S + 15) / 16;            // one wave32 per 16 segments
    const int gpb    = BT / 32;                  // 8 waves per block
    occl_kernel<<<(groups + gpb - 1) / gpb, BT, 0, stream>>>(
        outFull, triVec, segHit, S, TP);

    finalize_kernel<<<(N + BT - 1) / BT, BT, 0, stream>>>(mask12, segHit, outMask, N);
}